// QCNNi__59407987638934
// MI455X (gfx1250) — compile-verified
//
#include <hip/hip_runtime.h>
#include <hip/hip_bf16.h>

// ---------------------------------------------------------------------------
// QCNN: conv(4x4,s2) feature extraction via V_WMMA_F32_16X16X4_F32, then a
// 12-qubit statevector simulation with the 32KB state resident in LDS
// (interleaved complex, b64/b128 DS traffic). One workgroup per batch
// element; fused rotation gates; async global->LDS feature load; tree-
// reduced <Z>.
// ---------------------------------------------------------------------------

typedef float v2f __attribute__((ext_vector_type(2)));
typedef float v8f __attribute__((ext_vector_type(8)));

#define NQ      12
#define DIM     4096        // 2^NQ amplitudes
#define CTHREADS 128        // circuit threads per block (4 waves)

// ---------------- Kernel 1: conv front-end as im2col GEMV on WMMA ----------
// rows m = b*36 + j (j = oy*6+ox), K = 16 patch taps, w replicated over N.
__global__ void __launch_bounds__(32)
qcnn_conv_wmma(const float* __restrict__ x, const float* __restrict__ w,
               const float* __restrict__ bias, float* __restrict__ f,
               int nrows) {
  const int tile = blockIdx.x;
  const int lane = threadIdx.x;     // 0..31, one full wave
  const int half = lane >> 4;       // 0: K pair (0,1); 1: K pair (2,3)
  const int l    = lane & 15;       // M row within tile

  const int row   = tile * 16 + l;
  const bool valid = row < nrows;
  const int rsafe = valid ? row : 0;
  const int b  = rsafe / 36;
  const int j  = rsafe % 36;
  const int oy = j / 6, ox = j % 6;
  const float* xb = x + b * 196 + (2 * oy) * 14 + (2 * ox);

  v8f acc = {};                     // C starts at zero; bias added at store
  #pragma unroll
  for (int c = 0; c < 4; ++c) {
    // A 16x4 f32 layout: VGPR0 = K{4c+0 | 4c+2}, VGPR1 = K{4c+1 | 4c+3}
    const int k0 = 4 * c + (half ? 2 : 0);
    const int k1 = k0 + 1;
    v2f a, bb;
    a.x = valid ? xb[(k0 >> 2) * 14 + (k0 & 3)] : 0.0f;
    a.y = valid ? xb[(k1 >> 2) * 14 + (k1 & 3)] : 0.0f;
    bb.x = w[k0];                   // B row K replicated across all 16 N cols
    bb.y = w[k1];
    acc = __builtin_amdgcn_wmma_f32_16x16x4_f32(
        /*neg_a=*/false, a, /*neg_b=*/false, bb,
        /*c_mod=*/(short)0, acc, /*reuse_a=*/false, /*reuse_b=*/false);
  }

  // D layout: VGPR r -> (M=r, N=lane) lanes0-15 ; (M=8+r, N=lane-16) lanes16-31
  // Every N column is identical, so lanes with N==0 (lane 0 and 16) store.
  const float bv = bias[0];
  if (l == 0) {
    const int mbase = tile * 16 + half * 8;
    #pragma unroll
    for (int r = 0; r < 8; ++r) {
      const int m = mbase + r;
      if (m < nrows) f[m] = acc[r] + bv;
    }
  }
}

// ---------------- Kernel 2: 12-qubit statevector in LDS --------------------
struct alignas(8) c2 { float re, im; };
__device__ __forceinline__ c2 cmul(c2 a, c2 b) {
  return { a.re * b.re - a.im * b.im, a.re * b.im + a.im * b.re };
}
__device__ __forceinline__ c2 cadd(c2 a, c2 b) { return { a.re + b.re, a.im + b.im }; }

__device__ __forceinline__ void mrx(float t, c2 U[2][2]) {
  float c = __cosf(0.5f * t), s = __sinf(0.5f * t);
  U[0][0] = { c, 0.f }; U[0][1] = { 0.f, -s };
  U[1][0] = { 0.f, -s }; U[1][1] = { c, 0.f };
}
__device__ __forceinline__ void mry(float t, c2 U[2][2]) {
  float c = __cosf(0.5f * t), s = __sinf(0.5f * t);
  U[0][0] = { c, 0.f }; U[0][1] = { -s, 0.f };
  U[1][0] = { s, 0.f }; U[1][1] = { c, 0.f };
}
__device__ __forceinline__ void mrz(float t, c2 U[2][2]) {
  float c = __cosf(0.5f * t), s = __sinf(0.5f * t);
  U[0][0] = { c, -s }; U[0][1] = { 0.f, 0.f };
  U[1][0] = { 0.f, 0.f }; U[1][1] = { c, s };
}
__device__ __forceinline__ void mm2(const c2 A[2][2], const c2 B[2][2], c2 C[2][2]) {
  #pragma unroll
  for (int i = 0; i < 2; ++i)
    #pragma unroll
    for (int j = 0; j < 2; ++j)
      C[i][j] = cadd(cmul(A[i][0], B[0][j]), cmul(A[i][1], B[1][j]));
}

// apply 2x2 complex gate on bit position bp (wire q <-> bit 11-q)
__device__ __forceinline__ void apply1q(c2* sv, const c2 U[2][2], int bp, int tid) {
  const int h = 1 << bp;
  for (int p = tid; p < DIM / 2; p += CTHREADS) {
    const int lo = ((p >> bp) << (bp + 1)) | (p & (h - 1));
    const int hi = lo | h;
    c2 a = sv[lo];                 // ds_load_b64
    c2 b = sv[hi];
    c2 n0 = cadd(cmul(U[0][0], a), cmul(U[0][1], b));
    c2 n1 = cadd(cmul(U[1][0], a), cmul(U[1][1], b));
    sv[lo] = n0;                   // ds_store_b64
    sv[hi] = n1;
  }
}

// iSWAP on bit positions bpA,bpB: amp(A=1,B=0) <-> i*amp(A=0,B=1)
__device__ __forceinline__ void applyISwap(c2* sv, int bpA, int bpB, int tid) {
  const int bl = bpA < bpB ? bpA : bpB;
  const int bh = bpA < bpB ? bpB : bpA;
  for (int p = tid; p < DIM / 4; p += CTHREADS) {
    const int t    = ((p >> bl) << (bl + 1)) | (p & ((1 << bl) - 1));
    const int base = ((t >> bh) << (bh + 1)) | (t & ((1 << bh) - 1));
    const int iA = base | (1 << bpA);
    const int iB = base | (1 << bpB);
    c2 x = sv[iA];
    c2 y = sv[iB];
    sv[iA] = { -y.im, y.re };   // i * y
    sv[iB] = { -x.im, x.re };   // i * x
  }
}

__global__ void __launch_bounds__(CTHREADS)
qcnn_circuit(const float* __restrict__ f, const float* __restrict__ theta,
             float* __restrict__ out) {
  __shared__ c2    sv[DIM];            // 32KB interleaved statevector
  __shared__ float sfeat[36];
  __shared__ float sred[2][CTHREADS];

  const int b = blockIdx.x;
  const int tid = threadIdx.x;

  __builtin_prefetch(f + b * 36, 0, 0);   // global_prefetch_b8

  // ---- feature row -> LDS (async global->LDS on gfx1250 when available) ---
#if defined(__gfx1250__) && __has_builtin(__builtin_amdgcn_global_load_async_to_lds_b32)
  {
    typedef __attribute__((address_space(1))) int gint;  // "__device__ int *"
    typedef __attribute__((address_space(3))) int lint;  // LDS int *
    if (tid < 36) {
      gint* gsrc = (gint*)(unsigned long long)(const void*)(f + b * 36 + tid);
      // flat LDS address: low 32 bits are the wave-relative LDS byte offset
      lint* ldst = (lint*)(unsigned)(unsigned long long)(void*)(sfeat + tid);
      __builtin_amdgcn_global_load_async_to_lds_b32(gsrc, ldst, 0, 0);
    }
  #if __has_builtin(__builtin_amdgcn_s_wait_asynccnt)
    __builtin_amdgcn_s_wait_asynccnt(0);
  #else
    asm volatile("s_wait_asynccnt 0x0" ::: "memory");
  #endif
  }
#else
  if (tid < 36) sfeat[tid] = f[b * 36 + tid];
#endif

  // ---- zero-init statevector with b128 stores, set |0...0> ----------------
  {
    float4* v4 = (float4*)sv;          // DIM/2 float4 chunks
    const float4 z4 = make_float4(0.f, 0.f, 0.f, 0.f);
    for (int i = tid; i < DIM / 2; i += CTHREADS) v4[i] = z4;
  }
  __syncthreads();
  if (tid == 0) sv[0] = { 1.f, 0.f };
  __syncthreads();

  c2 G0[2][2], G1[2][2], G2[2][2], T[2][2], U[2][2];

  // encoder1: RX(a0) then RY(a1) then RX(a2)  ->  U = RX(a2)·RY(a1)·RX(a0)
  for (int q = 0; q < NQ; ++q) {
    const float a0 = sfeat[q * 3], a1 = sfeat[q * 3 + 1], a2 = sfeat[q * 3 + 2];
    mrx(a0, G0); mry(a1, G1); mrx(a2, G2);
    mm2(G1, G0, T); mm2(G2, T, U);
    apply1q(sv, U, 11 - q, tid);
    __syncthreads();
  }
  // iswap ascending: wires (11-i, 10-i) -> bits (i, i+1)
  for (int i = 0; i < NQ - 1; ++i) {
    applyISwap(sv, i, i + 1, tid);
    __syncthreads();
  }
  // encoder2: RY(a0), RX(a1), RY(a2) -> U = RY(a2)·RX(a1)·RY(a0)
  for (int q = 0; q < NQ; ++q) {
    const float a0 = sfeat[q * 3], a1 = sfeat[q * 3 + 1], a2 = sfeat[q * 3 + 2];
    mry(a0, G0); mrx(a1, G1); mry(a2, G2);
    mm2(G1, G0, T); mm2(G2, T, U);
    apply1q(sv, U, 11 - q, tid);
    __syncthreads();
  }
  // iswap descending: wires (i+1, i) -> bits (10-i, 11-i)
  for (int i = 0; i < NQ - 1; ++i) {
    applyISwap(sv, 10 - i, 11 - i, tid);
    __syncthreads();
  }
  // trainable RX,RY,RZ (shared params) on wires 0 and 1 -> bits 11, 10
  mrx(theta[0], G0); mry(theta[1], G1); mrz(theta[2], G2);
  mm2(G1, G0, T); mm2(G2, T, U);
  apply1q(sv, U, 11, tid);
  __syncthreads();
  apply1q(sv, U, 10, tid);
  __syncthreads();

  // <Z> on wires 0 (bit 11) and 1 (bit 10)
  float z0 = 0.f, z1 = 0.f;
  for (int i = tid; i < DIM; i += CTHREADS) {
    const c2 a = sv[i];
    const float p = a.re * a.re + a.im * a.im;
    z0 += (i & (1 << 11)) ? -p : p;
    z1 += (i & (1 << 10)) ? -p : p;
  }
  sred[0][tid] = z0; sred[1][tid] = z1;
  __syncthreads();
  for (int s = CTHREADS / 2; s > 0; s >>= 1) {
    if (tid < s) {
      sred[0][tid] += sred[0][tid + s];
      sred[1][tid] += sred[1][tid + s];
    }
    __syncthreads();
  }
  if (tid == 0) {
    out[b * 2 + 0] = sred[0][0];
    out[b * 2 + 1] = sred[1][0];
  }
}

// ---------------------------------------------------------------------------
extern "C" void kernel_launch(void* const* d_in, const int* in_sizes, int n_in,
                              void* d_out, int out_size, void* d_ws, size_t ws_size,
                              hipStream_t stream) {
  const float* x  = (const float*)d_in[0];   // [B,1,14,14]
  const float* cw = (const float*)d_in[1];   // [1,1,4,4]
  const float* cb = (const float*)d_in[2];   // [1]
  const float* th = (const float*)d_in[3];   // [3]

  const int B     = in_sizes[0] / 196;
  const int nrows = B * 36;
  const int tiles = (nrows + 15) / 16;

  float* feats = (float*)d_ws;               // B*36 floats of scratch

  qcnn_conv_wmma<<<tiles, 32, 0, stream>>>(x, cw, cb, feats, nrows);
  qcnn_circuit<<<B, CTHREADS, 0, stream>>>(feats, th, (float*)d_out);
}